// SolveSchedulingQP_69080253989322
// MI455X (gfx1250) — compile-verified
//
#include <hip/hip_runtime.h>
#include <stdint.h>

namespace {
constexpr int   kN     = 128;   // variables per problem
constexpr int   kNC    = 127;   // constraints per side (m = 2*kNC = 254)
constexpr int   kIters = 25;
constexpr float kH     = 0.4f;  // c_ramp
constexpr float kSigma = 0.1f;
constexpr float kBig   = 1e30f;
constexpr float kInvM  = 1.0f / 254.0f;
}

// gfx1250 async global->LDS copy (ASYNCcnt path). LDS operand is the per-lane
// LDS byte offset (low 32 bits of the generic shared-aperture address).
__device__ __forceinline__ void asyncLoadF32ToLds(float* ldsDst, const float* gSrc) {
  uint32_t loff = (uint32_t)(uintptr_t)ldsDst;
  unsigned long long gaddr = (unsigned long long)(uintptr_t)gSrc;
  asm volatile("global_load_async_to_lds_b32 %0, %1, off"
               :: "v"(loff), "v"(gaddr) : "memory");
}

__device__ __forceinline__ float waveSum(float v) {
#pragma unroll
  for (int m = 16; m >= 1; m >>= 1) v += __shfl_xor(v, m, 32);
  return v;
}
__device__ __forceinline__ float waveMin(float v) {
#pragma unroll
  for (int m = 16; m >= 1; m >>= 1) v = fminf(v, __shfl_xor(v, m, 32));
  return v;
}

// 128-thread (4-wave) block reductions, wave32-aware.
// NOTE: no trailing barrier. The scratch array is written once per PDIPM
// iteration and its next write is separated from this read by >=10 barriers,
// so the read-before-rewrite hazard cannot occur. The internal barrier also
// publishes any LDS writes made by all threads before the call.
__device__ __forceinline__ float blockSum(float v, float* red) {
  int t = threadIdx.x;
  v = waveSum(v);
  if ((t & 31) == 0) red[t >> 5] = v;
  __syncthreads();
  return (red[0] + red[1]) + (red[2] + red[3]);
}
__device__ __forceinline__ float blockMin(float v, float* red) {
  int t = threadIdx.x;
  v = waveMin(v);
  if ((t & 31) == 0) red[t >> 5] = v;
  __syncthreads();
  return fminf(fminf(red[0], red[1]), fminf(red[2], red[3]));
}

// PCR ghost equation: out-of-range rows behave as identity (b=1, rest 0).
__device__ __forceinline__ void ldEq(const float (*buf)[kN], int j,
                                     float& a, float& b, float& c, float& d) {
  if (j >= 0 && j < kN) { a = buf[0][j]; b = buf[1][j]; c = buf[2][j]; d = buf[3][j]; }
  else                  { a = 0.f; b = 1.f; c = 0.f; d = 0.f; }
}

__global__ __launch_bounds__(kN)
void pdipm_tridiag_kernel(const float* __restrict__ gZ0, const float* __restrict__ gMu,
                          const float* __restrict__ gDg, const float* __restrict__ gD2g,
                          float* __restrict__ gOut) {
  const int  t = threadIdx.x;
  const int  b = blockIdx.x;
  const bool c = (t < kNC);           // owns a constraint pair

  __shared__ float in0[kN], in1[kN], in2[kN], in3[kN];
  __shared__ float zs[kN], l1s[kN], l2s[kN], dzs[kN];
  __shared__ float u1s[kN], v1s[kN], u2s[kN], v2s[kN], ws[kN];
  __shared__ float bufA[4][kN], bufB[4][kN];   // PCR ping-pong: a,b,c,d rows
  __shared__ float redS[4], redM[4];

  const int base = b * kN + t;
  // Async-stage the four input rows through LDS (gfx1250 ASYNC path).
  asyncLoadF32ToLds(&in0[t], gZ0  + base);
  asyncLoadF32ToLds(&in1[t], gMu  + base);
  asyncLoadF32ToLds(&in2[t], gDg  + base);
  asyncLoadF32ToLds(&in3[t], gD2g + base);
  asm volatile("s_wait_asynccnt 0" ::: "memory");
  __syncthreads();

  const float z0v = in0[t], muv = in1[t], dgv = in2[t], d2gv = in3[t];
  const float Qd = d2gv + 1.0f;                 // diag of Q
  const float pv = dgv - d2gv * z0v - muv;      // linear term

  // State: z (all t), slack/dual pairs (t < 127). z=0 strictly feasible.
  float zreg = 0.f;
  float s1 = kH, s2 = kH, l1 = 1.f, l2 = 1.f;
  zs[t]  = 0.f;
  l1s[t] = c ? 1.f : 0.f;
  l2s[t] = c ? 1.f : 0.f;
  __syncthreads();

  for (int it = 0; it < kIters; ++it) {
    // Gz row t: z[t]-z[t+1] (and its negation for the second block).
    float zn  = c ? zs[t + 1] : 0.f;
    float gz  = zreg - zn;
    float rp1 = gz + s1 - kH;                   // primal residual, block 1
    float rp2 = -gz + s2 - kH;                  // primal residual, block 2

    // (G' lam)[t] = (l1[t]-l1[t-1]) - (l2[t]-l2[t-1]) with zero padding.
    float l1m = (t > 0) ? l1s[t - 1] : 0.f;
    float l2m = (t > 0) ? l2s[t - 1] : 0.f;
    float l1c = c ? l1 : 0.f;
    float l2c = c ? l2 : 0.f;
    float rd  = Qd * zreg + pv + (l1c - l1m) - (l2c - l2m);   // dual residual

    // t1 = d1*rhs2_1 is linear in mu:  t1 = u1 - mu*v1.
    // Publish u,v,w before the mu reduction so its barrier covers them.
    float d1 = 0.f, d2 = 0.f;
    if (c) {
      d1 = l1 / s1; d2 = l2 / s2;
      u1s[t] = d1 * (s1 - rp1);
      v1s[t] = kSigma * d1 / l1;
      u2s[t] = d2 * (s2 - rp2);
      v2s[t] = kSigma * d2 / l2;
      ws[t]  = d1 + d2;                         // tridiagonal weight
    } else {
      u1s[t] = 0.f; v1s[t] = 0.f; u2s[t] = 0.f; v2s[t] = 0.f; ws[t] = 0.f;
    }

    // Complementarity mu = mean(s*lam) over m=254 (barrier inside).
    float cm = blockSum(c ? (s1 * l1 + s2 * l2) : 0.f, redS) * kInvM;

    // rhs = -r_d + G'(d*rhs2);  M = diag(Qd) + D' diag(w) D  (tridiagonal).
    float wm  = (t > 0) ? ws[t - 1] : 0.f;
    float wc  = ws[t];                          // ws[127] == 0
    float t1c = u1s[t] - cm * v1s[t];
    float t2c = u2s[t] - cm * v2s[t];
    float t1m = (t > 0) ? (u1s[t - 1] - cm * v1s[t - 1]) : 0.f;
    float t2m = (t > 0) ? (u2s[t - 1] - cm * v2s[t - 1]) : 0.f;
    float rhs = -rd + (t1c - t1m) - (t2c - t2m);

    bufA[0][t] = -wm;             // sub-diagonal
    bufA[1][t] = Qd + wm + wc;    // diagonal (SPD, strictly dominant)
    bufA[2][t] = -wc;             // super-diagonal
    bufA[3][t] = rhs;
    __syncthreads();

    // Parallel cyclic reduction: 7 steps for n=128.
    float (*cur)[kN] = bufA;
    float (*alt)[kN] = bufB;
#pragma unroll
    for (int stp = 1; stp < kN; stp <<= 1) {
      float am, bm, cmm, dm, ap, bp, cp, dp;
      ldEq(cur, t - stp, am, bm, cmm, dm);
      ldEq(cur, t + stp, ap, bp, cp, dp);
      float ai = cur[0][t], bi = cur[1][t], ci = cur[2][t], di = cur[3][t];
      float fa = ai / bm;
      float fc = ci / bp;
      alt[0][t] = -fa * am;
      alt[1][t] = bi - fa * cmm - fc * ap;
      alt[2][t] = -fc * cp;
      alt[3][t] = di - fa * dm - fc * dp;
      __syncthreads();
      float (*tmp)[kN] = cur; cur = alt; alt = tmp;
    }
    float dz = cur[3][t] / cur[1][t];
    dzs[t] = dz;
    __syncthreads();

    // dlam = d * (G dz - rhs2); ds = (sigma*mu - s*lam - s*dlam)/lam.
    float dzn = c ? dzs[t + 1] : 0.f;
    float gdz = dz - dzn;
    float dl1 = 0.f, dl2 = 0.f, ds1 = 0.f, ds2 = 0.f, lim = kBig;
    if (c) {
      float r21 = -rp1 + s1 - kSigma * cm / l1;  // rhs2, block 1
      float r22 = -rp2 + s2 - kSigma * cm / l2;  // rhs2, block 2
      dl1 = d1 * (gdz - r21);
      dl2 = d2 * (-gdz - r22);
      ds1 = (kSigma * cm - s1 * l1 - s1 * dl1) / l1;
      ds2 = (kSigma * cm - s2 * l2 - s2 * dl2) / l2;
      lim = fminf(lim, ds1 < 0.f ? s1 / (-ds1) : kBig);
      lim = fminf(lim, ds2 < 0.f ? s2 / (-ds2) : kBig);
      lim = fminf(lim, dl1 < 0.f ? l1 / (-dl1) : kBig);
      lim = fminf(lim, dl2 < 0.f ? l2 / (-dl2) : kBig);
    }
    float alpha = fminf(1.0f, 0.99f * blockMin(lim, redM));

    zreg += alpha * dz;
    if (c) {
      s1 += alpha * ds1; s2 += alpha * ds2;
      l1 += alpha * dl1; l2 += alpha * dl2;
    }
    zs[t]  = zreg;
    l1s[t] = c ? l1 : 0.f;
    l2s[t] = c ? l2 : 0.f;
    __syncthreads();
  }

  __builtin_nontemporal_store(zreg, gOut + base);
}

extern "C" void kernel_launch(void* const* d_in, const int* in_sizes, int n_in,
                              void* d_out, int out_size, void* d_ws, size_t ws_size,
                              hipStream_t stream) {
  (void)n_in; (void)out_size; (void)d_ws; (void)ws_size;
  const float* z0  = (const float*)d_in[0];
  const float* mu  = (const float*)d_in[1];
  const float* dg  = (const float*)d_in[2];
  const float* d2g = (const float*)d_in[3];
  float* out = (float*)d_out;
  const int B = in_sizes[0] / kN;   // 512 batches
  pdipm_tridiag_kernel<<<B, kN, 0, stream>>>(z0, mu, dg, d2g, out);
}